// DAIN_4277787427373
// MI455X (gfx1250) — compile-verified
//
#include <hip/hip_runtime.h>
#include <cstdint>
#include <cstddef>

// Problem constants (fixed by the reference's setup_inputs).
#define B_   2
#define H_   192
#define W_   320
#define HW_  (H_ * W_)          // 61440
#define N_   (B_ * HW_)         // 122880 pixels total
#define CTX_ 195

// 16-byte vector with alignment lowered to 4 so the backend may emit a single
// (unaligned) global_load_b128 for a 4-float row segment.
typedef float f32x4 __attribute__((ext_vector_type(4), aligned(4)));

// ---------------------------------------------------------------------------
// Workspace zeroing (atomic accumulators must start at 0 every launch).
// ---------------------------------------------------------------------------
__global__ void k_zero(float* __restrict__ p, int n) {
    int i = blockIdx.x * blockDim.x + threadIdx.x;
    if (i < n) p[i] = 0.0f;
}

// ---------------------------------------------------------------------------
// depth_flow_project scatter: for both flow sets, splat (-f*d, d) onto the 4
// integer neighbors of (x+fx, y+fy). ws layout per set s:
//   accx = ws + s*3N, accy = +N, cnt = +2N
// ---------------------------------------------------------------------------
__global__ void k_splat(const float* __restrict__ flow01,
                        const float* __restrict__ flow10,
                        const float* __restrict__ d0,
                        const float* __restrict__ d1,
                        float* __restrict__ ws) {
    int gid = blockIdx.x * blockDim.x + threadIdx.x;
    if (gid >= 2 * N_) return;
    int s   = gid / N_;
    int p   = gid - s * N_;
    int b   = p / HW_;
    int rem = p - b * HW_;
    int y   = rem / W_;
    int x   = rem - y * W_;

    const float* fl = s ? flow10 : flow01;
    const float* dd = s ? d1 : d0;
    float fx = fl[(size_t)b * 2 * HW_ + rem];
    float fy = fl[(size_t)b * 2 * HW_ + HW_ + rem];
    float d  = dd[p];

    float x2 = (float)x + fx;
    float y2 = (float)y + fy;
    // If no neighbor can be in-bounds, all contributions are zero -> skip.
    // (Also guards the float->int cast against huge values.)
    if (!(x2 > -2.0f && x2 < (float)W_ + 1.0f &&
          y2 > -2.0f && y2 < (float)H_ + 1.0f)) return;

    int xL = (int)floorf(x2);
    int yT = (int)floorf(y2);

    float* base = ws + (size_t)s * 3 * N_;
    float* accx = base;
    float* accy = base + N_;
    float* cnt  = base + 2 * N_;
    float ax = -fx * d, ay = -fy * d;

#pragma unroll
    for (int dy = 0; dy < 2; ++dy) {
#pragma unroll
        for (int dx = 0; dx < 2; ++dx) {
            int xi = xL + dx, yi = yT + dy;
            if (xi >= 0 && xi < W_ && yi >= 0 && yi < H_) {
                int t = b * HW_ + yi * W_ + xi;
                atomicAdd(accx + t, ax);
                atomicAdd(accy + t, ay);
                atomicAdd(cnt  + t, d);
            }
        }
    }
}

// ---------------------------------------------------------------------------
// Normalize accumulators -> off0 / off1 (layout (B,2,H,W)), 0 where cnt==0.
// ---------------------------------------------------------------------------
__global__ void k_norm(const float* __restrict__ ws,
                       float* __restrict__ off0,
                       float* __restrict__ off1) {
    int gid = blockIdx.x * blockDim.x + threadIdx.x;
    if (gid >= 2 * N_) return;
    int s = gid / N_;
    int p = gid - s * N_;
    const float* base = ws + (size_t)s * 3 * N_;
    float c = base[2 * N_ + p];
    float ox = 0.0f, oy = 0.0f;
    if (c > 0.0f) {
        float inv = 1.0f / c;
        ox = base[p] * inv;
        oy = base[N_ + p] * inv;
    }
    int b   = p / HW_;
    int rem = p - b * HW_;
    float* o = s ? off1 : off0;
    o[(size_t)b * 2 * HW_ + rem]       = ox;
    o[(size_t)b * 2 * HW_ + HW_ + rem] = oy;
}

// ---------------------------------------------------------------------------
// filter_interpolate. One thread = one pixel x one channel-slice (blockIdx.y
// selects the slice; SLICES=4 for the 195-channel ctx warp -> 4x the waves in
// flight on this L2-gather-latency-bound kernel).
// The 16 filt planes for the block's 256 contiguous pixels are async-DMA'd
// into LDS (global_load_async_to_lds_b32, ASYNCcnt). Interior pixels collapse
// the 64 bilinear corners into a 5x5 window with 25 channel-independent
// weights; edge pixels use the exact reference clip-then-clip 16-tap path.
// Output stores are non-temporal to keep ctx0/ctx2 resident in the 192MB L2.
// ---------------------------------------------------------------------------
template <int C, int SLICES>
__global__ __launch_bounds__(256)
void k_filter(const float* __restrict__ src,
              const float* __restrict__ off,
              const float* __restrict__ filt,
              float* __restrict__ dst) {
    __shared__ float sfilt[16 * 256];   // 16 KB of 320 KB WGP LDS

    const int tid = threadIdx.x;
    const int pix = blockIdx.x * 256 + tid;   // HW_ % 256 == 0 -> b uniform
    const int b   = pix / HW_;
    const int rem = pix - b * HW_;
    const int y   = rem / W_;
    const int x   = rem - y * W_;

    constexpr int CPT = (C + SLICES - 1) / SLICES;
    const int c0 = blockIdx.y * CPT;
    const int c1 = (c0 + CPT < C) ? (c0 + CPT) : C;

    // --- async stage the 16 coalesced filt planes into LDS -----------------
    {
        unsigned long long g =
            (unsigned long long)(uintptr_t)(filt + (size_t)b * 16 * HW_ + rem);
        unsigned lds = (unsigned)(uintptr_t)(&sfilt[tid]);  // low 32 bits = LDS byte offset
#pragma unroll
        for (int t = 0; t < 16; ++t) {
            asm volatile("global_load_async_to_lds_b32 %0, %1, off"
                         :: "v"(lds), "v"(g) : "memory");
            g   += (unsigned long long)HW_ * 4ull;
            lds += 256u * 4u;
        }
        asm volatile("s_wait_asynccnt 0" ::: "memory");
    }

    float ox = off[(size_t)b * 2 * HW_ + rem];
    float oy = off[(size_t)b * 2 * HW_ + HW_ + rem];
    float x2 = (float)x + ox;
    float y2 = (float)y + oy;
    bool valid = (x2 >= 0.0f) && (x2 <= (float)(W_ - 1)) &&
                 (y2 >= 0.0f) && (y2 <= (float)(H_ - 1));

    float* d = dst + ((size_t)b * C + c0) * HW_ + rem;
    if (!valid) {
        for (int c = c0; c < c1; ++c) {
            __builtin_nontemporal_store(0.0f, d);
            d += HW_;
        }
        return;
    }

    float xLf = floorf(x2), yTf = floorf(y2);
    float aa = x2 - xLf, bb = y2 - yTf;
    float ia = 1.0f - aa, ib = 1.0f - bb;
    int xL = (int)xLf, yT = (int)yTf;

    float ft[16];
#pragma unroll
    for (int t = 0; t < 16; ++t) ft[t] = sfilt[t * 256 + tid];

    const bool interior = (xL >= 1) && (xL <= W_ - 4) && (yT >= 1) && (yT <= H_ - 4);

    if (interior) {
        // Tap fi's left corner lands on window column dx=fi (weight 1-a),
        // right corner on dx=fi+1 (weight a); same in y with fj / b.
        float Wg[25];
#pragma unroll
        for (int dy = 0; dy < 5; ++dy) {
#pragma unroll
            for (int dx = 0; dx < 5; ++dx) {
                float w = 0.0f;
#pragma unroll
                for (int fj = dy - 1; fj <= dy; ++fj) {
                    if (fj < 0 || fj > 3) continue;
                    float yw = (fj == dy) ? ib : bb;
#pragma unroll
                    for (int fi = dx - 1; fi <= dx; ++fi) {
                        if (fi < 0 || fi > 3) continue;
                        float xw = (fi == dx) ? ia : aa;
                        w = fmaf(ft[fj * 4 + fi], xw * yw, w);
                    }
                }
                Wg[dy * 5 + dx] = w;
            }
        }
        const float* p = src + ((size_t)b * C + c0) * HW_
                             + (size_t)(yT - 1) * W_ + (xL - 1);
        for (int c = c0; c < c1; ++c) {
            if (c + 2 < c1) {   // hide L2 latency two channels ahead
                const float* q = p + 2 * (size_t)HW_;
#pragma unroll
                for (int dy = 0; dy < 5; ++dy) __builtin_prefetch(q + dy * W_, 0, 1);
            }
            float acc = 0.0f;
#pragma unroll
            for (int dy = 0; dy < 5; ++dy) {
                const float* r = p + dy * W_;
                f32x4 v = *(const f32x4*)r;      // dx = 0..3 (one b128 if unaligned ok)
                float e = r[4];                  // dx = 4
                acc = fmaf(Wg[dy * 5 + 0], v.x, acc);
                acc = fmaf(Wg[dy * 5 + 1], v.y, acc);
                acc = fmaf(Wg[dy * 5 + 2], v.z, acc);
                acc = fmaf(Wg[dy * 5 + 3], v.w, acc);
                acc = fmaf(Wg[dy * 5 + 4], e,   acc);
            }
            __builtin_nontemporal_store(acc, d);
            d += HW_;
            p += HW_;
        }
    } else {
        // Exact reference edge behavior: clip corner, then clip corner+1.
        int xs[4], xs1[4], ys[4], ys1[4];
#pragma unroll
        for (int f = 0; f < 4; ++f) {
            int xi = min(max(xL + f - 1, 0), W_ - 1);
            int yi = min(max(yT + f - 1, 0), H_ - 1);
            xs[f]  = xi;  xs1[f] = min(xi + 1, W_ - 1);
            ys[f]  = yi;  ys1[f] = min(yi + 1, H_ - 1);
        }
        const float w00 = ia * ib, w10 = aa * ib, w01 = ia * bb, w11 = aa * bb;
        const float* p = src + ((size_t)b * C + c0) * HW_;
        for (int c = c0; c < c1; ++c) {
            float acc = 0.0f;
#pragma unroll
            for (int fj = 0; fj < 4; ++fj) {
                const float* rT = p + ys[fj]  * W_;
                const float* rB = p + ys1[fj] * W_;
#pragma unroll
                for (int fi = 0; fi < 4; ++fi) {
                    float tl = rT[xs[fi]],  tr = rT[xs1[fi]];
                    float bl = rB[xs[fi]],  br = rB[xs1[fi]];
                    float samp = w00 * tl + w10 * tr + w01 * bl + w11 * br;
                    acc = fmaf(ft[fj * 4 + fi], samp, acc);
                }
            }
            __builtin_nontemporal_store(acc, d);
            d += HW_;
            p += HW_;
        }
    }
}

// ---------------------------------------------------------------------------
// cur_output = ref0/2 + ref2/2
// ---------------------------------------------------------------------------
__global__ void k_combine(const float* __restrict__ r0,
                          const float* __restrict__ r2,
                          float* __restrict__ cur, int n) {
    int i = blockIdx.x * blockDim.x + threadIdx.x;
    if (i < n) __builtin_nontemporal_store(0.5f * (r0[i] + r2[i]), cur + i);
}

// ---------------------------------------------------------------------------
extern "C" void kernel_launch(void* const* d_in, const int* in_sizes, int n_in,
                              void* d_out, int out_size, void* d_ws, size_t ws_size,
                              hipStream_t stream) {
    (void)in_sizes; (void)n_in; (void)out_size; (void)ws_size;

    const float* im0    = (const float*)d_in[0];
    const float* im2    = (const float*)d_in[1];
    const float* ctx0   = (const float*)d_in[2];
    const float* ctx2   = (const float*)d_in[3];
    const float* flow01 = (const float*)d_in[4];
    const float* flow10 = (const float*)d_in[5];
    const float* dep0   = (const float*)d_in[6];
    const float* dep1   = (const float*)d_in[7];
    const float* filt0  = (const float*)d_in[8];
    const float* filt1  = (const float*)d_in[9];

    float* out   = (float*)d_out;
    float* cur   = out;                              // (B,3,H,W)
    float* ref0  = cur  + (size_t)3 * N_;            // (B,3,H,W)
    float* ref2  = ref0 + (size_t)3 * N_;            // (B,3,H,W)
    float* ctx0w = ref2 + (size_t)3 * N_;            // (B,195,H,W)
    float* ctx2w = ctx0w + (size_t)CTX_ * N_;        // (B,195,H,W)
    float* off0  = ctx2w + (size_t)CTX_ * N_;        // (B,2,H,W)
    float* off1  = off0 + (size_t)2 * N_;            // (B,2,H,W)

    float* ws = (float*)d_ws;                        // 2 * 3 * N_ floats (~2.9 MB)

    const int zn = 2 * 3 * N_;
    k_zero<<<(zn + 255) / 256, 256, 0, stream>>>(ws, zn);
    k_splat<<<(2 * N_ + 255) / 256, 256, 0, stream>>>(flow01, flow10, dep0, dep1, ws);
    k_norm<<<(2 * N_ + 255) / 256, 256, 0, stream>>>(ws, off0, off1);

    dim3 fg3(N_ / 256, 1);   // 480 blocks x 256 threads (8 wave32 per block)
    k_filter<3, 1><<<fg3, 256, 0, stream>>>(im0, off0, filt0, ref0);
    k_filter<3, 1><<<fg3, 256, 0, stream>>>(im2, off1, filt1, ref2);

    dim3 fgc(N_ / 256, 4);   // 1920 blocks -> 15360 waves for the ctx gathers
    k_filter<CTX_, 4><<<fgc, 256, 0, stream>>>(ctx0, off0, filt0, ctx0w);
    k_filter<CTX_, 4><<<fgc, 256, 0, stream>>>(ctx2, off1, filt1, ctx2w);

    const int cn = 3 * N_;
    k_combine<<<(cn + 255) / 256, 256, 0, stream>>>(ref0, ref2, cur, cn);
}